// Net_53687091200141
// MI455X (gfx1250) — compile-verified
//
#include <hip/hip_runtime.h>
#include <hip/hip_bf16.h>

typedef __attribute__((ext_vector_type(2))) float v2f;
typedef __attribute__((ext_vector_type(8))) float v8f;

// ---------------------------------------------------------------------------
// init: deg[i] = 1.0 (accounts for the self-loop edge), zero the layer-1
// aggregation buffer.
// ---------------------------------------------------------------------------
__global__ void gcn_init_kernel(float* __restrict__ deg, float* __restrict__ hagg, int n) {
    int i = blockIdx.x * blockDim.x + threadIdx.x;
    if (i < n * 16) hagg[i] = 0.0f;
    if (i < n)      deg[i]  = 1.0f;   // self-loop contribution to degree
}

// ---------------------------------------------------------------------------
// degree accumulation over targets (col)
// ---------------------------------------------------------------------------
__global__ void gcn_deg_kernel(float* __restrict__ deg, const int* __restrict__ col, int nE) {
    int e = blockIdx.x * blockDim.x + threadIdx.x;
    if (e < nE) atomicAdd(&deg[col[e]], 1.0f);
}

// ---------------------------------------------------------------------------
// dinv = deg > 0 ? rsqrt(deg) : 0   (in place)
// ---------------------------------------------------------------------------
__global__ void gcn_rsqrt_kernel(float* __restrict__ d, int n) {
    int i = blockIdx.x * blockDim.x + threadIdx.x;
    if (i < n) {
        float v = d[i];
        d[i] = (v > 0.0f) ? rsqrtf(v) : 0.0f;
    }
}

// ---------------------------------------------------------------------------
// xw = x @ W1 via V_WMMA_F32_16X16X4_F32 (K=2 padded to 4 with zeros).
// One wave per 16 rows.
//   A (16x4 f32):  lanes 0-15 hold M=lane, VGPR0=K0, VGPR1=K1;
//                  lanes 16-31 hold K2/K3 -> zero padding.
//   B (4x16):      lane holds N=lane&15, VGPRv holds K=v+2*(lane>=16);
//                  rows K=2,3 -> zero padding.
//   C (16x16):     VGPR r, lane L -> (M = r + 8*(L>>4), N = L&15).
// Loads are unconditional (valid addresses) + v_cndmask masking; stores are
// branch-free coalesced b32 rows in the (common) full-tile case.
// ---------------------------------------------------------------------------
__global__ void gcn_xw_wmma_kernel(const float* __restrict__ x,
                                   const float* __restrict__ W1,
                                   float* __restrict__ xw, int n) {
    int wave = (blockIdx.x * blockDim.x + threadIdx.x) >> 5;
    int lane = threadIdx.x & 31;
    int base = wave * 16;
    if (base >= n) return;              // wave-uniform -> EXEC all-1 at WMMA

    bool lo = (lane < 16);
    int  nc = lane & 15;
    int  m  = base + nc;
    int  mc = (m < n) ? m : (n - 1);    // clamped: always a valid address

    // Unconditional loads, then mask with cndmask (no exec-mask branches).
    float x0 = x[2 * mc + 0];
    float x1 = x[2 * mc + 1];
    float w0 = W1[0 * 16 + nc];
    float w1 = W1[1 * 16 + nc];

    bool av = lo && (m < n);
    v2f a, b;
    a.x = av ? x0 : 0.0f;               // K=0 (K=2 pad -> 0)
    a.y = av ? x1 : 0.0f;               // K=1 (K=3 pad -> 0)
    b.x = lo ? w0 : 0.0f;               // B row K=0 (K=2 pad -> 0)
    b.y = lo ? w1 : 0.0f;               // B row K=1 (K=3 pad -> 0)

    v8f c = {};
    c = __builtin_amdgcn_wmma_f32_16x16x4_f32(
            /*neg_a=*/false, a, /*neg_b=*/false, b,
            /*c_mod=*/(short)0, c, /*reuse_a=*/false, /*reuse_b=*/false);

    int half = lane >> 4;
    float* dst = xw + (size_t)(base + 8 * half) * 16 + nc;
    if (base + 16 <= n) {
        // Full tile (always true when n % 16 == 0): straight-line coalesced
        // stores, row r at byte offset r*64.
#pragma unroll
        for (int r = 0; r < 8; ++r) dst[(size_t)r * 16] = c[r];
    } else {
        int rows = n - base;            // partial last tile
#pragma unroll
        for (int r = 0; r < 8; ++r)
            if (r + 8 * half < rows) dst[(size_t)r * 16] = c[r];
    }
}

// ---------------------------------------------------------------------------
// layer-1 edge pass: 16 lanes per edge (one per feature) -> coalesced gather
// of xw[row] and coalesced atomic scatter into hagg[col].
// ---------------------------------------------------------------------------
__global__ void gcn_edge1_kernel(const float* __restrict__ xw,
                                 const float* __restrict__ dinv,
                                 const int* __restrict__ row,
                                 const int* __restrict__ col,
                                 float* __restrict__ hagg, int nE) {
    unsigned tid = blockIdx.x * blockDim.x + threadIdx.x;
    unsigned e   = tid >> 4;
    if (e >= (unsigned)nE) return;
    int j = tid & 15;
    int r = row[e];
    int c = col[e];
    float w = dinv[r] * dinv[c];
    atomicAdd(&hagg[(size_t)c * 16 + j], xw[(size_t)r * 16 + j] * w);
}

// ---------------------------------------------------------------------------
// fused node pass:
//   h[i,j] = relu(hagg[i,j] + xw[i,j]*dinv[i]^2 + b1[j])   (self-loop folded in)
//   hw[i]  = h[i,:] . W2
//   out[i] = b2 + hw[i]*dinv[i]^2                          (layer-2 self-loop + bias)
// ---------------------------------------------------------------------------
__global__ void gcn_fuse_kernel(const float* __restrict__ hagg,
                                const float* __restrict__ xw,
                                const float* __restrict__ dinv,
                                const float* __restrict__ b1,
                                const float* __restrict__ W2,
                                const float* __restrict__ b2,
                                float* __restrict__ hw,
                                float* __restrict__ out, int n) {
    int i = blockIdx.x * blockDim.x + threadIdx.x;
    if (i >= n) return;
    float di = dinv[i];
    float sl = di * di;
    const float4* ha = (const float4*)(hagg + (size_t)i * 16);
    const float4* xv = (const float4*)(xw   + (size_t)i * 16);
    const float4* bb = (const float4*)b1;
    const float4* ww = (const float4*)W2;
    float acc = 0.0f;
#pragma unroll
    for (int q = 0; q < 4; ++q) {
        float4 h4 = ha[q];
        float4 x4 = xv[q];
        float4 b4 = bb[q];
        float4 w4 = ww[q];
        float v0 = fmaxf(fmaf(x4.x, sl, h4.x) + b4.x, 0.0f);
        float v1 = fmaxf(fmaf(x4.y, sl, h4.y) + b4.y, 0.0f);
        float v2 = fmaxf(fmaf(x4.z, sl, h4.z) + b4.z, 0.0f);
        float v3 = fmaxf(fmaf(x4.w, sl, h4.w) + b4.w, 0.0f);
        acc = fmaf(v0, w4.x, acc);
        acc = fmaf(v1, w4.y, acc);
        acc = fmaf(v2, w4.z, acc);
        acc = fmaf(v3, w4.w, acc);
    }
    hw[i]  = acc;
    out[i] = b2[0] + acc * sl;
}

// ---------------------------------------------------------------------------
// layer-2 edge pass: scalar gather + atomic scatter
// ---------------------------------------------------------------------------
__global__ void gcn_edge2_kernel(const float* __restrict__ hw,
                                 const float* __restrict__ dinv,
                                 const int* __restrict__ row,
                                 const int* __restrict__ col,
                                 float* __restrict__ out, int nE) {
    int e = blockIdx.x * blockDim.x + threadIdx.x;
    if (e >= nE) return;
    int r = row[e];
    int c = col[e];
    atomicAdd(&out[c], hw[r] * dinv[r] * dinv[c]);
}

extern "C" void kernel_launch(void* const* d_in, const int* in_sizes, int n_in,
                              void* d_out, int out_size, void* d_ws, size_t ws_size,
                              hipStream_t stream) {
    const float* x  = (const float*)d_in[0];   // [n,2]
    const int*   ei = (const int*)  d_in[1];   // [2,nE]
    const float* W1 = (const float*)d_in[2];   // [2,16]
    const float* b1 = (const float*)d_in[3];   // [16]
    const float* W2 = (const float*)d_in[4];   // [16,1]
    const float* b2 = (const float*)d_in[5];   // [1]

    const int n  = in_sizes[0] / 2;
    const int nE = in_sizes[1] / 2;
    const int* row = ei;        // edge_index[0] : sources
    const int* col = ei + nE;   // edge_index[1] : targets

    float* ws   = (float*)d_ws;
    float* dinv = ws;                       // n     (deg -> dinv in place)
    float* xw   = ws + (size_t)n;           // 16n
    float* hagg = ws + (size_t)17 * n;      // 16n
    float* hw   = ws + (size_t)33 * n;      // n

    float* out = (float*)d_out;

    const int T = 256;
    // 1) init deg (self-loop) + zero hagg
    gcn_init_kernel<<<(n * 16 + T - 1) / T, T, 0, stream>>>(dinv, hagg, n);
    // 2) degree over targets
    gcn_deg_kernel<<<(nE + T - 1) / T, T, 0, stream>>>(dinv, col, nE);
    // 3) dinv = rsqrt(deg)
    gcn_rsqrt_kernel<<<(n + T - 1) / T, T, 0, stream>>>(dinv, n);
    // 4) xw = x @ W1 via WMMA f32 16x16x4 (one wave per 16 rows)
    {
        int waves  = (n + 15) / 16;
        int blocks = (waves + 3) / 4;       // 128 threads = 4 waves per block
        gcn_xw_wmma_kernel<<<blocks, 128, 0, stream>>>(x, W1, xw, n);
    }
    // 5) layer-1 aggregation: 16 lanes per edge
    {
        size_t total    = (size_t)nE * 16;
        unsigned blocks = (unsigned)((total + T - 1) / T);
        gcn_edge1_kernel<<<blocks, T, 0, stream>>>(xw, dinv, row, col, hagg, nE);
    }
    // 6) fused self-loop + bias + relu + @W2 + layer-2 self-loop + b2
    gcn_fuse_kernel<<<(n + T - 1) / T, T, 0, stream>>>(hagg, xw, dinv, b1, W2, b2, hw, out, n);
    // 7) layer-2 aggregation
    gcn_edge2_kernel<<<(nE + T - 1) / T, T, 0, stream>>>(hw, dinv, row, col, out, nE);
}